// EWMALayer_32710470926987
// MI455X (gfx1250) — compile-verified
//
#include <hip/hip_runtime.h>
#include <cstdint>

typedef __attribute__((ext_vector_type(4))) float v4f;
typedef __attribute__((ext_vector_type(4))) int   v4i;

#define EW_ALPHA   0.3f
#define TSTEPS     4096
#define NCH        16384                 // 64*256 channels
#define BLOCK      256                   // 8 wave32s
#define CH_PER_BLOCK (BLOCK * 4)         // 1024 channels per block (float4/lane)
#define NGROUP     (NCH / CH_PER_BLOCK)  // 16 channel groups
#define NCHUNK     16                    // time chunks (parallel via warm-up)
#define CHUNK_LEN  (TSTEPS / NCHUNK)     // 256
#define WARMUP     64                    // 0.7^64 ~ 1.2e-10 << fp32 eps; multiple of 4
#define DEPTH      16                    // async-to-LDS ring depth (power of 2)
#define ROWV       (NCH / 4)             // float4 columns per timestep = 4096

#if defined(__gfx1250__) && __has_builtin(__builtin_amdgcn_global_load_async_to_lds_b128)
#define USE_ASYNC 1
#else
#define USE_ASYNC 0
#endif

typedef __attribute__((address_space(1))) v4i g_v4i;
typedef __attribute__((address_space(3))) v4i l_v4i;

__global__ __launch_bounds__(BLOCK)
void ewma_scan_kernel(const float* __restrict__ xg, float* __restrict__ yg)
{
    const int tid   = threadIdx.x;
    const int group = blockIdx.x;        // channel group
    const int chunk = blockIdx.y;        // time chunk

    const int tstart = chunk * CHUNK_LEN;
    const int t0     = (chunk == 0) ? 0 : (tstart - WARMUP);
    const int tend   = tstart + CHUNK_LEN;
    const int nsteps = tend - t0;        // 256 or 320; multiple of DEPTH and 4
    const int nwarm  = tstart - t0;      // 0 or 64; multiple of 4

    const int cvec = group * (CH_PER_BLOCK / 4) + tid;   // this thread's float4 column

    const v4f* __restrict__ xs = (const v4f*)xg + (size_t)t0     * ROWV + cvec;
    v4f*       __restrict__ ys = (v4f*)yg       + (size_t)tstart * ROWV + cvec;

    const float a = EW_ALPHA;
    const float b = 1.0f - EW_ALPHA;

    v4f yv;

#if USE_ASYNC
    // Per-thread 16-deep staging ring: 16 slots * 256 threads * 16B = 64 KB LDS.
    __shared__ v4f stage[DEPTH][BLOCK];

    // Re-arm helper: slot <- timestep (relative to t0).
    auto fill = [&](int slot, int step) {
        __builtin_amdgcn_global_load_async_to_lds_b128(
            (g_v4i*)(xs + (size_t)step * ROWV),
            (l_v4i*)&stage[slot][tid], 0, 0);
    };

    // Prologue: put DEPTH timesteps in flight.
    #pragma unroll
    for (int s = 0; s < DEPTH; ++s) fill(s, s);

    for (int i = 0; i < nsteps; i += 4) {
        // Async loads complete in order: outstanding <= DEPTH-4 => slots i..i+3 ready.
        asm volatile("s_wait_asynccnt %0" :: "n"(DEPTH - 4) : "memory");
        v4f x0 = stage[(i + 0) & (DEPTH - 1)][tid];
        v4f x1 = stage[(i + 1) & (DEPTH - 1)][tid];
        v4f x2 = stage[(i + 2) & (DEPTH - 1)][tid];
        v4f x3 = stage[(i + 3) & (DEPTH - 1)][tid];
        // Retire the ds_loads before re-arming these slots with async fills.
        asm volatile("s_wait_dscnt 0" ::: "memory");
        if (i + DEPTH < nsteps) {        // all-or-nothing: both sides multiples of 4
            fill((i + 0) & (DEPTH - 1), i + DEPTH + 0);
            fill((i + 1) & (DEPTH - 1), i + DEPTH + 1);
            fill((i + 2) & (DEPTH - 1), i + DEPTH + 2);
            fill((i + 3) & (DEPTH - 1), i + DEPTH + 3);
        }
        // Serial recurrence: y[t] = a*x[t] + (1-a)*y[t-1]; y[0] = x[0].
        v4f y0 = (i == 0) ? x0 : (x0 * a + yv * b);
        v4f y1 = x1 * a + y0 * b;
        v4f y2 = x2 * a + y1 * b;
        v4f y3 = x3 * a + y2 * b;
        yv = y3;
        if (i >= nwarm) {                // uniform scalar branch (nwarm multiple of 4)
            v4f* yo = ys + (size_t)(i - nwarm) * ROWV;
            __builtin_nontemporal_store(y0, yo + (size_t)0 * ROWV);
            __builtin_nontemporal_store(y1, yo + (size_t)1 * ROWV);
            __builtin_nontemporal_store(y2, yo + (size_t)2 * ROWV);
            __builtin_nontemporal_store(y3, yo + (size_t)3 * ROWV);
        }
    }
    asm volatile("s_wait_asynccnt 0" ::: "memory");
#else
    // Fallback: direct vector loads with software prefetch (emits global_prefetch_b8).
    yv = xs[0];
    if (0 >= nwarm) __builtin_nontemporal_store(yv, ys);
    for (int i = 1; i < nsteps; ++i) {
        if (i + 8 < nsteps) __builtin_prefetch(xs + (size_t)(i + 8) * ROWV, 0, 0);
        v4f xv = xs[(size_t)i * ROWV];
        yv = xv * a + yv * b;
        if (i >= nwarm) {
            __builtin_nontemporal_store(yv, ys + (size_t)(i - nwarm) * ROWV);
        }
    }
#endif
}

extern "C" void kernel_launch(void* const* d_in, const int* in_sizes, int n_in,
                              void* d_out, int out_size, void* d_ws, size_t ws_size,
                              hipStream_t stream) {
    const float* x = (const float*)d_in[0];
    float*       y = (float*)d_out;
    (void)in_sizes; (void)n_in; (void)out_size; (void)d_ws; (void)ws_size;

    dim3 grid(NGROUP, NCHUNK);   // 16 channel groups x 16 time chunks = 256 blocks
    ewma_scan_kernel<<<grid, BLOCK, 0, stream>>>(x, y);
}